// DigiCapsLayer_13357348291338
// MI455X (gfx1250) — compile-verified
//
#include <hip/hip_runtime.h>
#include <math.h>

typedef __attribute__((ext_vector_type(2))) float v2f;
typedef __attribute__((ext_vector_type(8))) float v8f;

#define BATCH    4096
#define NIN      1152
#define DIN      8
#define DOUT     16
#define KCAPS    10
#define NITER    3
#define SU_STRIDE 17          // pad to avoid LDS bank conflicts on column reads
#define CS_STRIDE 16
#define THREADS  256
#define NWAVES   8
#define CHUNKS   (NIN / (NWAVES * 4))   // 36 WMMA k-steps per wave for s-einsum
#define TILES_W  ((NIN / 16) / NWAVES)  // 9 row-tiles per wave for bs-update

// shared memory partition (floats)
#define OFF_U     0
#define OFF_BS    (OFF_U  + NIN * SU_STRIDE)       // 19584
#define OFF_CS    (OFF_BS + NIN * KCAPS)           // +11520
#define OFF_RED   (OFF_CS + NIN * CS_STRIDE)       // +18432
#define OFF_S     (OFF_RED + NWAVES * 256)         // +2048
#define OFF_V     (OFF_S  + 256)
#define OFF_PART  (OFF_V  + 256)
#define OFF_MX    (OFF_PART + 256)
#define OFF_RSUM  (OFF_MX + 16)
#define OFF_W     (OFF_RSUM + 16)
#define OFF_BIAS  (OFF_W + DOUT * DIN)
#define SMEM_FLOATS (OFF_BIAS + 16)                // 52528 floats = 210112 B < 320 KB

__global__ __launch_bounds__(THREADS)
void digicaps_routing_kernel(const float* __restrict__ uin,
                             const float* __restrict__ Wg,
                             const float* __restrict__ bg,
                             float* __restrict__ out) {
  extern __shared__ float smem[];
  float* sU    = smem + OFF_U;     // [NIN][SU_STRIDE] projected capsules
  float* sBs   = smem + OFF_BS;    // [NIN][KCAPS] routing logits
  float* sCs   = smem + OFF_CS;    // [NIN][16] unnormalized softmax weights (cols>=10 zero)
  float* sRed  = smem + OFF_RED;   // per-wave partial C matrices
  float* sS    = smem + OFF_S;     // [16][16] s matrix
  float* sV    = smem + OFF_V;     // [16][16] v matrix (cols >= KCAPS zeroed)
  float* sPart = smem + OFF_PART;  // softmax reduction scratch
  float* sMx   = smem + OFF_MX;
  float* sRSum = smem + OFF_RSUM;
  float* sW    = smem + OFF_W;     // [DOUT][DIN]
  float* sBias = smem + OFF_BIAS;

  const int tid    = threadIdx.x;
  const int b      = blockIdx.x;
  const int waveId = tid >> 5;
  const int lane   = tid & 31;
  const int lhalf  = lane & 15;
  const int hi     = lane >> 4;    // 0 = lanes 0-15, 1 = lanes 16-31

  // ---- load W / bias, zero logits and cs (cols >= KCAPS stay zero forever) ----
  if (tid < DOUT * DIN) sW[tid] = Wg[tid];
  if (tid < DOUT)       sBias[tid] = bg[tid];
  for (int i = tid; i < NIN * KCAPS; i += THREADS) sBs[i] = 0.0f;
  for (int i = tid; i < NIN * CS_STRIDE; i += THREADS) sCs[i] = 0.0f;
  __syncthreads();

  // ---- phase 1: u = u_in @ W^T + bias  (VALU; 576 FMAs/thread) ----
  const float4* uin4 = (const float4*)uin + (size_t)b * NIN * 2;
  for (int n = tid; n < NIN; n += THREADS) {
    float4 p0 = uin4[n * 2 + 0];
    float4 p1 = uin4[n * 2 + 1];
    float in[8] = {p0.x, p0.y, p0.z, p0.w, p1.x, p1.y, p1.z, p1.w};
#pragma unroll
    for (int d = 0; d < DOUT; ++d) {
      float acc = sBias[d];
#pragma unroll
      for (int i = 0; i < DIN; ++i) acc = fmaf(in[i], sW[d * DIN + i], acc);
      sU[n * SU_STRIDE + d] = acc;
    }
  }
  __syncthreads();

  // ---- routing iterations, all in LDS ----
  for (int it = 0; it < NITER; ++it) {
    // softmax over n (per k): max pass
    {
      int k = tid & 15, g = tid >> 4;
      float m = -3.402823466e38f;
      if (k < KCAPS)
        for (int n = g; n < NIN; n += 16) m = fmaxf(m, sBs[n * KCAPS + k]);
      sPart[tid] = m;
    }
    __syncthreads();
    if (tid < KCAPS) {
      float m = sPart[tid];
#pragma unroll
      for (int g = 1; g < 16; ++g) m = fmaxf(m, sPart[g * 16 + tid]);
      sMx[tid] = m;
    }
    __syncthreads();
    // sum pass: also materializes e = exp(bs - mx) into sCs (unnormalized cs)
    {
      int k = tid & 15, g = tid >> 4;
      float ssum = 0.0f;
      if (k < KCAPS) {
        float m = sMx[k];
        for (int n = g; n < NIN; n += 16) {
          float e = __expf(sBs[n * KCAPS + k] - m);
          sCs[n * CS_STRIDE + k] = e;
          ssum += e;
        }
      }
      sPart[tid] = ssum;
    }
    __syncthreads();
    if (tid < 16) {
      float r = 0.0f;
      if (tid < KCAPS) {
        float ssum = 0.0f;
#pragma unroll
        for (int g = 0; g < 16; ++g) ssum += sPart[g * 16 + tid];
        r = 1.0f / ssum;
      }
      sRSum[tid] = r;   // zero for padded columns
    }
    __syncthreads();

    // ---- s[d][k] = (sum_n u[n][d] * e[n][k]) * rsum[k] : chained WMMA f32 16x16x4 ----
    // Clean inner body: 2x ds_load_2addr + v_wmma, no divergence, no transcendentals.
    v8f cacc = {0.f, 0.f, 0.f, 0.f, 0.f, 0.f, 0.f, 0.f};
    const int nBase = waveId * (NIN / NWAVES);
#pragma unroll 4
    for (int c4 = 0; c4 < CHUNKS; ++c4) {
      int n0 = nBase + c4 * 4 + 2 * hi;       // lane layout: VGPR0->K=2*hi, VGPR1->K=2*hi+1
      v2f a, bv;
      a.x  = sU[n0 * SU_STRIDE + lhalf];
      a.y  = sU[(n0 + 1) * SU_STRIDE + lhalf];
      bv.x = sCs[n0 * CS_STRIDE + lhalf];
      bv.y = sCs[(n0 + 1) * CS_STRIDE + lhalf];
      cacc = __builtin_amdgcn_wmma_f32_16x16x4_f32(
          false, a, false, bv, (short)0, cacc, false, false);
    }
    // cross-wave reduction of partial C matrices; fold in softmax denominator
#pragma unroll
    for (int r = 0; r < 8; ++r) sRed[waveId * 256 + r * 32 + lane] = cacc[r];
    __syncthreads();
    {
      float acc = 0.0f;
#pragma unroll
      for (int w = 0; w < NWAVES; ++w) acc += sRed[w * 256 + tid];
      int rr = tid >> 5, ll = tid & 31;
      int d = rr + 8 * (ll >> 4), k = ll & 15;   // C/D layout: VGPR r -> M=r / M=r+8
      sS[d * 16 + k] = acc * sRSum[k];
    }
    __syncthreads();

    // ---- squash over k (last axis, faithful to reference) ----
    if (tid < DOUT) {
      float nsq = 0.0f;
#pragma unroll
      for (int k = 0; k < KCAPS; ++k) { float x = sS[tid * 16 + k]; nsq = fmaf(x, x, nsq); }
      float nrm = sqrtf(nsq);
      float scal = (nsq / (1.0f + nrm)) / (1.0f + nsq) / nrm;
#pragma unroll
      for (int k = 0; k < 16; ++k)
        sV[tid * 16 + k] = (k < KCAPS) ? sS[tid * 16 + k] * scal : 0.0f;
    }
    __syncthreads();

    // ---- bs[n][k] += sum_d u[n][d] * v[d][k] : WMMA, 16-row tiles ----
    if (it < NITER - 1) {
      for (int tt = 0; tt < TILES_W; ++tt) {
        int n0 = (waveId * TILES_W + tt) * 16;
        v8f cc;
#pragma unroll
        for (int r = 0; r < 8; ++r)
          cc[r] = (lhalf < KCAPS) ? sBs[(n0 + r + 8 * hi) * KCAPS + lhalf] : 0.0f;
#pragma unroll
        for (int d0 = 0; d0 < DOUT; d0 += 4) {
          int dd = d0 + 2 * hi;
          v2f a, bv;
          a.x  = sU[(n0 + lhalf) * SU_STRIDE + dd];
          a.y  = sU[(n0 + lhalf) * SU_STRIDE + dd + 1];
          bv.x = sV[dd * 16 + lhalf];
          bv.y = sV[(dd + 1) * 16 + lhalf];
          cc = __builtin_amdgcn_wmma_f32_16x16x4_f32(
              false, a, false, bv, (short)0, cc, false, false);
        }
#pragma unroll
        for (int r = 0; r < 8; ++r)
          if (lhalf < KCAPS) sBs[(n0 + r + 8 * hi) * KCAPS + lhalf] = cc[r];
      }
      __syncthreads();
    }
  }

  // ---- output: out[b][k][d] = v[d][k] ----
  if (tid < KCAPS * DOUT) {
    int k = tid >> 4, d = tid & 15;
    out[((size_t)b * KCAPS + k) * DOUT + d] = sV[d * 16 + k];
  }
}

extern "C" void kernel_launch(void* const* d_in, const int* in_sizes, int n_in,
                              void* d_out, int out_size, void* d_ws, size_t ws_size,
                              hipStream_t stream) {
  (void)in_sizes; (void)n_in; (void)d_ws; (void)ws_size; (void)out_size;
  const float* u    = (const float*)d_in[0];
  const float* W    = (const float*)d_in[1];
  const float* bias = (const float*)d_in[2];
  float* out = (float*)d_out;

  const size_t smemBytes = (size_t)SMEM_FLOATS * sizeof(float); // ~205 KB < 320 KB
  hipFuncSetAttribute((const void*)digicaps_routing_kernel,
                      hipFuncAttributeMaxDynamicSharedMemorySize, (int)smemBytes);
  digicaps_routing_kernel<<<BATCH, THREADS, smemBytes, stream>>>(u, W, bias, out);
}